// TemporalCorrelation_19069654794413
// MI455X (gfx1250) — compile-verified
//
#include <hip/hip_runtime.h>

// ---- problem constants (from reference) ----
#define N_    4
#define C_    256
#define T_    16
#define H_    56
#define W_    56
#define G_    8      // C / GROUP_SIZE
#define CPG   32     // GROUP_SIZE (channels per group) == WMMA K
#define TAPS  9      // 3x3 kernel taps

// ---- tiling ----
#define HT      4            // H rows per block
#define HB      (H_/HT)      // 14 blocks along H
#define HALO_R  (HT+2)       // 6 rows of x(t) with halo
#define HALO_W  (W_+2)       // 58 cols with halo
#define NPIX    (HT*W_)      // 224 pixels per block strip
#define NGRP    (NPIX/16)    // 14 groups of 16 pixels
#define NUNITS  (TAPS*NGRP)  // 126 (tap, pixel-group) WMMA units per block

#define CSTR    (T_*H_*W_)   // channel stride in x (50176)

// ---- LDS layout (floats); x tiles stored channel-PAIRED: [c/2][r][w][2] ----
#define XH_SZ   (CPG*HALO_R*HALO_W)   // 11136 floats
#define XP_SZ   (CPG*HT*W_)           // 7168 floats
#define WK_SZ   (TAPS*CPG)            // 288 floats
#define LDS_FLOATS (XH_SZ + XP_SZ + WK_SZ)

typedef float f2  __attribute__((ext_vector_type(2)));
typedef float v8f __attribute__((ext_vector_type(8)));

__global__ __launch_bounds__(256) void
temporal_corr_wmma(const float* __restrict__ x,
                   const float* __restrict__ wgt,
                   float* __restrict__ out)
{
    extern __shared__ float lds[];
    float* xh = lds;                    // x(t) halo tile, paired: f2[CPG/2][HALO_R][HALO_W]
    float* xp = lds + XH_SZ;            // x(t-1) tile,   paired: f2[CPG/2][HT][W_]
    float* wk = lds + XH_SZ + XP_SZ;    // weights [TAPS][CPG] (c-contiguous -> natural pairs)

    const int bh  = blockIdx.x;         // H strip (0..13)
    const int t   = blockIdx.y;         // 0..15
    const int ng  = blockIdx.z;         // n*8 + g
    const int n   = ng >> 3;
    const int g   = ng & 7;
    const int tid = threadIdx.x;

    const int tprev = (t == 0) ? 0 : (t - 1);
    const int h0    = bh * HT;

    const float* xt  = x + ((long)(n*C_ + g*CPG) * T_ + t)     * (long)(H_*W_);
    const float* xtp = x + ((long)(n*C_ + g*CPG) * T_ + tprev) * (long)(H_*W_);

    // ---- stage x(t) halo tile (zero-padded), channel-paired layout ----
    for (int i = tid; i < XH_SZ; i += 256) {
        int c   = i / (HALO_R*HALO_W);
        int rem = i - c * (HALO_R*HALO_W);     // r*HALO_W + cc
        int r   = rem / HALO_W;
        int cc  = rem - r * HALO_W;
        int hh  = h0 - 1 + r;
        int ww  = cc - 1;
        float v = 0.0f;
        if (hh >= 0 && hh < H_ && ww >= 0 && ww < W_)
            v = xt[(long)c * CSTR + hh * W_ + ww];
        // paired store: f2 index = (c/2)*(HALO_R*HALO_W) + rem, component = c&1
        xh[(((c >> 1) * (HALO_R*HALO_W) + rem) << 1) + (c & 1)] = v;
    }
    // ---- stage x(t-1) tile, channel-paired layout ----
    for (int i = tid; i < XP_SZ; i += 256) {
        int c   = i / (HT*W_);
        int rem = i - c * (HT*W_);             // r*W_ + ww
        int r   = rem / W_;
        int ww  = rem - r * W_;
        float v = xtp[(long)c * CSTR + (h0 + r) * W_ + ww];
        xp[(((c >> 1) * (HT*W_) + rem) << 1) + (c & 1)] = v;
    }
    // ---- stage weights w[k][c] for this (g, t) ----
    for (int i = tid; i < WK_SZ; i += 256) {
        int k = i / CPG;
        int c = i - k * CPG;
        wk[i] = wgt[(long)(g*CPG + c) * (T_*TAPS) + t * TAPS + k];
    }
    __syncthreads();

    const int lane = tid & 31;
    const int wave = tid >> 5;
    const int half = lane >> 4;   // wave32 half: selects channel pairs {4j,4j+1} vs {4j+2,4j+3}
    const int n16  = lane & 15;   // pixel slot within the 16-wide N tile

    // Each unit: one tap k, one group of 16 pixels.
    //   out[p] = Sum_c w[c,k] * xprev[c,p] * xshift_k[c,p]
    // A(16x4) = w rows replicated; B(4x16) = per-pixel products; K=32 via 8 chained WMMAs.
    for (int u = wave; u < NUNITS; u += 8) {
        const int k   = u / NGRP;
        const int gi  = u - k * NGRP;
        const int kh  = k / 3;
        const int kw  = k - kh * 3;

        const int p   = gi * 16 + n16;     // 0..223
        const int row = p / W_;
        const int col = p - row * W_;

        const f2* xp2 = ((const f2*)xp) + row * W_ + col;                    // + c2*(HT*W_)
        const f2* xh2 = ((const f2*)xh) + (row + kh) * HALO_W + (col + kw);  // + c2*(HALO_R*HALO_W)
        const f2* wk2 = ((const f2*)wk) + k * (CPG/2);                       // + 2j + half

        // ---- batch all LDS loads (ds_load_b64), then compute ----
        f2 bp[8], bs[8], aw[8];
#pragma unroll
        for (int j = 0; j < 8; ++j) {
            const int c2 = 2*j + half;        // channel pair index (4j+2*half)/2
            bp[j] = xp2[c2 * (HT*W_)];
            bs[j] = xh2[c2 * (HALO_R*HALO_W)];
            aw[j] = wk2[c2];
        }

        // ---- two independent accumulator chains for WMMA pipelining ----
        v8f acc0 = {}, acc1 = {};
#pragma unroll
        for (int j = 0; j < 8; j += 2) {
            f2 y0 = bp[j]   * bs[j];
            f2 y1 = bp[j+1] * bs[j+1];
            acc0 = __builtin_amdgcn_wmma_f32_16x16x4_f32(
                     false, aw[j],   false, y0, (short)0, acc0, false, false);
            acc1 = __builtin_amdgcn_wmma_f32_16x16x4_f32(
                     false, aw[j+1], false, y1, (short)0, acc1, false, false);
        }

        // Every D row is identical; lane l holds D[.., N=l&15] in component 0.
        if (half == 0) {
            const size_t o = ((size_t)((n*G_ + g)*TAPS + k) * T_ + t) * (size_t)(H_*W_)
                           + (size_t)h0 * W_ + p;
            out[o] = acc0[0] + acc1[0];
        }
    }
}

extern "C" void kernel_launch(void* const* d_in, const int* in_sizes, int n_in,
                              void* d_out, int out_size, void* d_ws, size_t ws_size,
                              hipStream_t stream)
{
    const float* x   = (const float*)d_in[0];   // (4,256,16,56,56) fp32
    const float* wgt = (const float*)d_in[1];   // (256,16,3,3) fp32
    float* out = (float*)d_out;                 // (4,72,16,56,56) fp32

    dim3 grid(HB, T_, N_ * G_);                 // (14, 16, 32)
    dim3 block(256);                            // 8 wave32
    size_t shmem = LDS_FLOATS * sizeof(float);  // ~74 KB of the 320 KB WGP LDS

    temporal_corr_wmma<<<grid, block, shmem, stream>>>(x, wgt, out);
}